// Net_35983236006103
// MI455X (gfx1250) — compile-verified
//
#include <hip/hip_runtime.h>
#include <hip/hip_bf16.h>

typedef __attribute__((ext_vector_type(2))) float v2f;
typedef __attribute__((ext_vector_type(8))) float v8f;
typedef int v4i __attribute__((vector_size(4 * sizeof(int))));   // matches builtin proto

#define WPB 4   // waves per 128-thread block

// --- CDNA5 async global->LDS path (probe via __has_builtin; fallback is the
// --- known-good b128 load + ds_store staging).
#if defined(__has_builtin)
# if __has_builtin(__builtin_amdgcn_global_load_async_to_lds_b128) && \
     __has_builtin(__builtin_amdgcn_s_wait_asynccnt)
#  define USE_ASYNC_LDS 1
# endif
#endif
#ifndef USE_ASYNC_LDS
# define USE_ASYNC_LDS 0
#endif

#define AS1 __attribute__((address_space(1)))
#define AS3 __attribute__((address_space(3)))

static __device__ __forceinline__ v8f wmma_f32_k4(v2f a, v2f b, v8f c) {
    // V_WMMA_F32_16X16X4_F32: D = A(16x4) * B(4x16) + C(16x16), f32 throughout
    return __builtin_amdgcn_wmma_f32_16x16x4_f32(false, a, false, b, (short)0, c,
                                                 false, false);
}

// Stage one 28x28 fp32 image (196 float4) into a wave-private LDS slot.
static __device__ __forceinline__ void stage_image(const float* __restrict__ img,
                                                   float* dst, int lane) {
#if USE_ASYNC_LDS
    #pragma unroll
    for (int i = 0; i < 7; ++i) {
        int e = lane + 32 * i;                        // float4 index, 196 total
        if (e < 196) {
            __builtin_amdgcn_global_load_async_to_lds_b128(
                (AS1 v4i*)(img + (size_t)e * 4),
                (AS3 v4i*)(dst + (size_t)e * 4),
                0, 0);
        }
    }
#else
    #pragma unroll
    for (int i = 0; i < 7; ++i) {
        int e = lane + 32 * i;
        if (e < 196) {
            float4 vv = ((const float4*)img)[e];
            *(float4*)&dst[e * 4] = vv;
        }
    }
#endif
}

__global__ __launch_bounds__(128) void mlp_wmma_kernel(
    const float* __restrict__ x,
    const float* __restrict__ W1, const float* __restrict__ b1,
    const float* __restrict__ W2, const float* __restrict__ b2,
    const float* __restrict__ W3, const float* __restrict__ b3,
    float* __restrict__ out, int B)
{
    // Wave-private LDS slots: no cross-wave sharing -> zero barriers.
    __shared__ __align__(16) float sImg[WPB][2][800];   // double-buffered image
    __shared__ float sH[WPB][4][16][17];                // layer1 out: 4 samples x 16x16
    __shared__ float sY[WPB][16][66];                   // relu(layer2) flat: 16 x 64
    __shared__ float sZ[WPB][16][17];                   // logits: 16 x (10 of 16)

    const int tid   = threadIdx.x;
    const int lane  = tid & 31;
    const int wv    = tid >> 5;
    const int nIdx  = lane & 15;   // N column owned by this lane (B/C/D layouts)
    const int khalf = lane >> 4;   // 0: K{0,1}; 1: K{2,3} (A/B layouts)
    const int patch = lane & 15;   // layer-1 A row (patch index)
    const int pi    = patch >> 2;
    const int pj    = patch & 3;

    // ---- Preload all B-matrices (weights, transposed into 4x16 B layout) into VGPRs.
    // B[k][n] = W[n][k]; lane holds (k = 4t + 2*khalf + {0,1}, n = nIdx).
    float B1r[13][2];
    #pragma unroll
    for (int t = 0; t < 13; ++t) {
        #pragma unroll
        for (int v = 0; v < 2; ++v) {
            int k = 4 * t + 2 * khalf + v;
            B1r[t][v] = (k < 49) ? W1[nIdx * 49 + k] : 0.0f;   // K padded 49 -> 52
        }
    }
    float B2r[16][2];
    #pragma unroll
    for (int t = 0; t < 16; ++t) {
        #pragma unroll
        for (int v = 0; v < 2; ++v) {
            int k = 4 * t + 2 * khalf + v;           // k in [0,64)
            B2r[t][v] = W2[nIdx * 64 + k];
        }
    }
    float B3r[16][2];
    #pragma unroll
    for (int t = 0; t < 16; ++t) {
        #pragma unroll
        for (int v = 0; v < 2; ++v) {
            int k = 4 * t + 2 * khalf + v;
            B3r[t][v] = (nIdx < 10) ? W3[nIdx * 64 + k] : 0.0f;  // N padded 10 -> 16
        }
    }
    const float bias1 = b1[nIdx];
    const float bias2 = b2[nIdx];
    const float bias3 = (nIdx < 10) ? b3[nIdx] : 0.0f;

    const int       gw   = blockIdx.x * WPB + wv;           // global wave id
    const long long step = (long long)(gridDim.x * WPB) * 16;
    const long long base0 = (long long)gw * 16;

    // Prime the pipeline: prefetch image 0 of the first macro into buffer 0.
    if (base0 < (long long)B) {
        stage_image(x + (size_t)base0 * 784, &sImg[wv][0][0], lane);
    }

    for (long long base = base0; base < (long long)B; base += step) {

        #pragma unroll 1
        for (int g = 0; g < 4; ++g) {              // 4 subgroups of 4 samples
            // ================= Layer 1: per-sample 16x52x16 GEMM =================
            #pragma unroll 1
            for (int ss = 0; ss < 4; ++ss) {
                const int idx = g * 4 + ss;        // sample index within macro, 0..15
                const int buf = idx & 1;

                // Prefetch the NEXT image (crossing the macro boundary at idx==15)
                // so exactly 14 async ops are in flight; waiting to <=7 then
                // guarantees the current buffer is complete (in-order returns).
                long long nxt = (idx == 15) ? (base + step) : (base + idx + 1);
                if (nxt >= (long long)B) nxt = (long long)B - 1;   // clamped dup, harmless
                stage_image(x + (size_t)nxt * 784, &sImg[wv][buf ^ 1][0], lane);
#if USE_ASYNC_LDS
                __builtin_amdgcn_s_wait_asynccnt(7);
#endif
                const float* ib = &sImg[wv][buf][0];

                v8f acc = {0.f, 0.f, 0.f, 0.f, 0.f, 0.f, 0.f, 0.f};
                #pragma unroll
                for (int t = 0; t < 13; ++t) {
                    int k0 = 4 * t + 2 * khalf;
                    float a0 = 0.0f, a1 = 0.0f;
                    if (k0 < 49) {
                        int r = k0 / 7, c = k0 - 7 * r;
                        a0 = ib[(7 * pi + r) * 28 + 7 * pj + c];
                    }
                    if (k0 + 1 < 49) {
                        int k1 = k0 + 1;
                        int r = k1 / 7, c = k1 - 7 * r;
                        a1 = ib[(7 * pi + r) * 28 + 7 * pj + c];
                    }
                    v2f a = {a0, a1};
                    v2f bb = {B1r[t][0], B1r[t][1]};
                    acc = wmma_f32_k4(a, bb, acc);
                }
                // C/D layout: VGPR v holds row M = v + 8*khalf, col nIdx.
                #pragma unroll
                for (int v = 0; v < 8; ++v) {
                    int m = v + 8 * khalf;                     // patch row
                    sH[wv][ss][m][nIdx] = acc[v] + bias1;
                }
            }

            // ====== Layer 2: one 16x64x16 tile, rows = (sample-in-subgroup, block) ======
            const int r2  = lane & 15;      // A row: 4*ss + bl
            const int ss2 = r2 >> 2;
            const int ba  = (r2 & 3) >> 1;  // block coords (a, c)
            const int bc  = r2 & 1;

            v8f acc2 = {0.f, 0.f, 0.f, 0.f, 0.f, 0.f, 0.f, 0.f};
            #pragma unroll
            for (int t = 0; t < 16; ++t) {
                float av[2];
                #pragma unroll
                for (int v = 0; v < 2; ++v) {
                    int k  = 4 * t + 2 * khalf + v;            // k in [0,64)
                    int q  = k >> 4;                           // which patch of the block
                    int f  = k & 15;                           // feature
                    int qb = q >> 1, qd = q & 1;
                    int p2 = (2 * ba + qb) * 4 + (2 * bc + qd);
                    av[v] = sH[wv][ss2][p2][f];                // the 'hb' regroup
                }
                v2f a = {av[0], av[1]};
                v2f bb = {B2r[t][0], B2r[t][1]};
                acc2 = wmma_f32_k4(a, bb, acc2);
            }
            #pragma unroll
            for (int v = 0; v < 8; ++v) {
                int m   = v + 8 * khalf;                       // row = 4*smp + blk
                int smp = m >> 2;
                int blk = m & 3;
                float yv = acc2[v] + bias2;
                yv = fmaxf(yv, 0.0f);                          // ReLU
                sY[wv][g * 4 + smp][blk * 16 + nIdx] = yv;
            }
        }

        // ============ Layer 3: one 16x64x16 tile, rows = 16 samples ============
        const int smp = lane & 15;
        v8f acc3 = {0.f, 0.f, 0.f, 0.f, 0.f, 0.f, 0.f, 0.f};
        #pragma unroll
        for (int t = 0; t < 16; ++t) {
            int k0 = 4 * t + 2 * khalf;
            v2f a = {sY[wv][smp][k0], sY[wv][smp][k0 + 1]};
            v2f bb = {B3r[t][0], B3r[t][1]};
            acc3 = wmma_f32_k4(a, bb, acc3);
        }
        #pragma unroll
        for (int v = 0; v < 8; ++v) {
            int m = v + 8 * khalf;
            sZ[wv][m][nIdx] = acc3[v] + bias3;
        }

        // ---------------- log_softmax + store (lanes 0..15, one sample each) ----
        if (lane < 16) {
            long long s = base + lane;
            if (s < (long long)B) {
                float z[10];
                float mx = -3.4e38f;
                #pragma unroll
                for (int c = 0; c < 10; ++c) {
                    z[c] = sZ[wv][lane][c];
                    mx = fmaxf(mx, z[c]);
                }
                float se = 0.0f;
                #pragma unroll
                for (int c = 0; c < 10; ++c) se += __expf(z[c] - mx);
                float lse = mx + __logf(se);
                float* o = out + (size_t)s * 10;
                #pragma unroll
                for (int c = 0; c < 10; ++c) o[c] = z[c] - lse;
            }
        }
    }
}

extern "C" void kernel_launch(void* const* d_in, const int* in_sizes, int n_in,
                              void* d_out, int out_size, void* d_ws, size_t ws_size,
                              hipStream_t stream) {
    const float* x  = (const float*)d_in[0];
    const float* W1 = (const float*)d_in[1];
    const float* b1 = (const float*)d_in[2];
    const float* W2 = (const float*)d_in[3];
    const float* b2 = (const float*)d_in[4];
    const float* W3 = (const float*)d_in[5];
    const float* b3 = (const float*)d_in[6];
    float* out = (float*)d_out;

    const int B = in_sizes[0] / 784;
    int macros = (B + 15) / 16;                 // 16 samples per wave-macro
    int blocks = (macros + WPB - 1) / WPB;
    if (blocks < 1) blocks = 1;
    if (blocks > 16384) blocks = 16384;         // grid-stride handles the rest

    mlp_wmma_kernel<<<blocks, 32 * WPB, 0, stream>>>(x, W1, b1, W2, b2, W3, b3,
                                                     out, B);
}